// LGA3_89395449299013
// MI455X (gfx1250) — compile-verified
//
#include <hip/hip_runtime.h>
#include <stdint.h>

// LGA3 fused: 3 cascaded per-pixel 5x5 local-aggregation passes in one
// kernel, all intermediates kept in LDS. Memory-bound: compulsory HBM
// traffic 186MB (~8us @ 23.3TB/s); fusion removes the 2x134MB t1/t2 round
// trips of the naive 3-kernel version. Staging uses CDNA5 async
// global->LDS b128 copies (ASYNCcnt) with 16B-aligned groups; regions are
// widened to x0-8 so every 4-float group is fully in- or out-of-image,
// keeping the zero-padding logic exact.

#define RAD   2
#define C_TOT 32
#define CG    16          // channels per workgroup (2 channel-groups)
#define H_    256
#define W_    512
#define TH    16
#define TW    16
#define NTAP  25
#define BLK   256

// x region: rows ty0-6 .. ty0+21 (28), cols tx0-8 .. tx0+23 (32, 16B aligned)
#define XROWS 28
#define XCOLS 32
// f region: rows ty0-4 .. ty0+19 (24), cols tx0-8 .. tx0+19 (28, 16B aligned)
#define FROWS 24
#define FCOLS 28
// t1 region: 24x24 at (-4,-4); t2 region: 20x20 at (-2,-2)
#define T1R   24
#define T2R   20

typedef int v4i __attribute__((vector_size(4 * sizeof(int))));
typedef __attribute__((address_space(1))) int  GasI;
typedef __attribute__((address_space(3))) int  LasI;
typedef __attribute__((address_space(1))) v4i  Gas4;
typedef __attribute__((address_space(3))) v4i  Las4;

__device__ __forceinline__ void async_copy_b128(const float* g, float* l) {
#if __has_builtin(__builtin_amdgcn_global_load_async_to_lds_b128)
  __builtin_amdgcn_global_load_async_to_lds_b128(
      (Gas4*)(uintptr_t)g, (Las4*)(uintptr_t)l, 0, 0);
#elif __has_builtin(__builtin_amdgcn_global_load_async_to_lds_b32)
  __builtin_amdgcn_global_load_async_to_lds_b32(
      (GasI*)(uintptr_t)(g + 0), (LasI*)(uintptr_t)(l + 0), 0, 0);
  __builtin_amdgcn_global_load_async_to_lds_b32(
      (GasI*)(uintptr_t)(g + 1), (LasI*)(uintptr_t)(l + 1), 0, 0);
  __builtin_amdgcn_global_load_async_to_lds_b32(
      (GasI*)(uintptr_t)(g + 2), (LasI*)(uintptr_t)(l + 2), 0, 0);
  __builtin_amdgcn_global_load_async_to_lds_b32(
      (GasI*)(uintptr_t)(g + 3), (LasI*)(uintptr_t)(l + 3), 0, 0);
#else
  unsigned lo = (unsigned)(uintptr_t)l;
  asm volatile("global_load_async_to_lds_b128 %0, %1, off"
               :: "v"(lo), "v"(g) : "memory");
#endif
}

__device__ __forceinline__ void wait_async0() {
#if __has_builtin(__builtin_amdgcn_s_wait_asynccnt)
  __builtin_amdgcn_s_wait_asynccnt(0);
#else
  asm volatile("s_wait_asynccnt 0" ::: "memory");
#endif
}

__global__ __launch_bounds__(BLK)
void lga3_fused(const float* __restrict__ x,
                const float* __restrict__ f,
                float* __restrict__ out)
{
  // 57344 B; reused as t2 (CG*20*20*4 = 25600 B fits)
  __shared__ __align__(16) float sX [CG * XROWS * XCOLS];
  // 36864 B
  __shared__ __align__(16) float sT1[CG * T1R * T1R];
  // 67200 B
  __shared__ __align__(16) float sF [NTAP * FROWS * FCOLS];
  // total 161408 B static LDS: 2 workgroups co-reside in a 320KB WGP

  const int tid = threadIdx.x;
  const int tx0 = blockIdx.x * TW;
  const int ty0 = blockIdx.y * TH;
  const int n   = blockIdx.z >> 1;
  const int cg  = blockIdx.z & 1;

  // ---------- stage x tile via async b128 (16B-aligned groups) ----------
  {
    const size_t xbase = ((size_t)n * C_TOT + cg * CG) * (size_t)(H_ * W_);
    const int NXG = CG * XROWS * (XCOLS / 4);          // 3584 groups
    for (int i = tid; i < NXG; i += BLK) {
      int c  = i / (XROWS * (XCOLS / 4));
      int r  = i - c * (XROWS * (XCOLS / 4));
      int ly = r >> 3, j = r & 7;                      // XCOLS/4 == 8
      int gy = ty0 - 6 + ly;
      int gx = tx0 - 8 + 4 * j;                        // 4-elem aligned group
      int li = c * (XROWS * XCOLS) + ly * XCOLS + 4 * j;
      if ((unsigned)gy < H_ && (unsigned)gx < W_) {    // group fully in image
        async_copy_b128(&x[xbase + (size_t)c * (H_ * W_) + gy * W_ + gx],
                        &sX[li]);
      } else {                                         // zero padding
        *(float4*)&sX[li] = make_float4(0.f, 0.f, 0.f, 0.f);
      }
    }
  }

  // ---------- stage f tile via async b128 ----------
  // f loaded as 0 outside the image => t1/t2 computed at out-of-image
  // positions are exactly 0, matching the reference's zero padding.
  {
    const size_t fbase = (size_t)n * NTAP * (size_t)(H_ * W_);
    const int NFG = NTAP * FROWS * (FCOLS / 4);        // 4200 groups
    for (int i = tid; i < NFG; i += BLK) {
      int k  = i / (FROWS * (FCOLS / 4));
      int r  = i - k * (FROWS * (FCOLS / 4));
      int ly = r / 7, j = r - ly * 7;                  // FCOLS/4 == 7
      int gy = ty0 - 4 + ly;
      int gx = tx0 - 8 + 4 * j;
      int li = k * (FROWS * FCOLS) + ly * FCOLS + 4 * j;
      if ((unsigned)gy < H_ && (unsigned)gx < W_) {
        async_copy_b128(&f[fbase + (size_t)k * (H_ * W_) + gy * W_ + gx],
                        &sF[li]);
      } else {
        *(float4*)&sF[li] = make_float4(0.f, 0.f, 0.f, 0.f);
      }
    }
  }

  wait_async0();        // ASYNCcnt == 0 : all async LDS writes landed
  __syncthreads();

  // ---------- pass 1: t1 over 24x24 ----------
  // t1(py,px) <-> global (ty0-4+py, tx0-4+px); x col = px+2+dx, f col = px+4
  for (int pos = tid; pos < T1R * T1R; pos += BLK) {
    int py = pos / T1R, px = pos - py * T1R;
    float fk[NTAP];
#pragma unroll
    for (int k = 0; k < NTAP; ++k)
      fk[k] = sF[k * (FROWS * FCOLS) + py * FCOLS + (px + 4)];
    for (int c = 0; c < CG; ++c) {
      const float* xb = &sX[c * (XROWS * XCOLS) + py * XCOLS + (px + 2)];
      float acc = 0.f;
#pragma unroll
      for (int dy = 0; dy < 5; ++dy)
#pragma unroll
        for (int dx = 0; dx < 5; ++dx)
          acc += fk[dy * 5 + dx] * xb[dy * XCOLS + dx];
      sT1[c * (T1R * T1R) + pos] = acc;
    }
  }
  __syncthreads();

  // ---------- pass 2: t2 over 20x20, written into the x buffer ----------
  // t2(py,px) <-> global (ty0-2+py, tx0-2+px); f at (py+2, px+6)
  float* sT2 = sX;   // x fully consumed after the barrier above
  for (int pos = tid; pos < T2R * T2R; pos += BLK) {
    int py = pos / T2R, px = pos - py * T2R;
    float fk[NTAP];
#pragma unroll
    for (int k = 0; k < NTAP; ++k)
      fk[k] = sF[k * (FROWS * FCOLS) + (py + 2) * FCOLS + (px + 6)];
    for (int c = 0; c < CG; ++c) {
      const float* tb = &sT1[c * (T1R * T1R) + py * T1R + px];
      float acc = 0.f;
#pragma unroll
      for (int dy = 0; dy < 5; ++dy)
#pragma unroll
        for (int dx = 0; dx < 5; ++dx)
          acc += fk[dy * 5 + dx] * tb[dy * T1R + dx];
      sT2[c * (T2R * T2R) + pos] = acc;
    }
  }
  __syncthreads();

  // ---------- pass 3: out 16x16, one pixel per thread, all channels ----------
  // out(py,px) <-> global (ty0+py, tx0+px); f at (py+4, px+8)
  {
    int py = tid >> 4, px = tid & 15;
    float fk[NTAP];
#pragma unroll
    for (int k = 0; k < NTAP; ++k)
      fk[k] = sF[k * (FROWS * FCOLS) + (py + 4) * FCOLS + (px + 8)];
    for (int c = 0; c < CG; ++c) {
      const float* tb = &sT2[c * (T2R * T2R) + py * T2R + px];
      float acc = 0.f;
#pragma unroll
      for (int dy = 0; dy < 5; ++dy)
#pragma unroll
        for (int dx = 0; dx < 5; ++dx)
          acc += fk[dy * 5 + dx] * tb[dy * T2R + dx];
      out[(((size_t)n * C_TOT + cg * CG + c) * H_ + ty0 + py) * W_ +
          tx0 + px] = acc;
    }
  }
}

extern "C" void kernel_launch(void* const* d_in, const int* in_sizes, int n_in,
                              void* d_out, int out_size, void* d_ws, size_t ws_size,
                              hipStream_t stream) {
  (void)n_in; (void)out_size; (void)d_ws; (void)ws_size;
  const float* x = (const float*)d_in[0];   // [N,32,256,512] f32
  const float* f = (const float*)d_in[1];   // [N,25,256,512] f32
  float* out     = (float*)d_out;           // [N,32,256,512] f32

  const int N = in_sizes[0] / (C_TOT * H_ * W_);
  dim3 grid(W_ / TW, H_ / TH, N * 2);       // 32 x 16 x 8 = 4096 WGs
  lga3_fused<<<grid, BLK, 0, stream>>>(x, f, out);
}